// Convolution_51213190038150
// MI455X (gfx1250) — compile-verified
//
#include <hip/hip_runtime.h>
#include <math.h>

// e3nn-style equivariant graph convolution for MI455X (gfx1250, wave32).
// Edge MLP (dominant 9.2 GFLOP fp32) runs on V_WMMA_F32_16X16X4_F32.
// Scatter (segment_sum) uses lane-coalesced global_atomic_add_f32 into an
// L2-resident (N,256) accumulator (51.2MB << 192MB L2).
// Workspace layout (floats): f0f1[N*128] | scx[N*128] | mid[N*256]  (~102.4MB)

typedef __attribute__((ext_vector_type(2))) float v2f;
typedef __attribute__((ext_vector_type(8))) float v8f;

__device__ __forceinline__ v8f wmma4(v2f a, v2f b, v8f c) {
  // D = A(16x4,f32) * B(4x16,f32) + C(16x16,f32)
  return __builtin_amdgcn_wmma_f32_16x16x4_f32(false, a, false, b, (short)0, c,
                                               false, false);
}

__global__ void zero_kernel(float* __restrict__ p, long n) {
  long i = (long)blockIdx.x * blockDim.x + threadIdx.x;
  long stride = (long)gridDim.x * blockDim.x;
  for (long k = i * 4; k < n; k += stride * 4) {
    *(float4*)(p + k) = make_float4(0.f, 0.f, 0.f, 0.f);
  }
}

// One wave per node; lane = output channel w (0..31).
// f0f1 layout: [f0(32) | f1_m0(32) | f1_m1(32) | f1_m2(32)]  (planes for coalesced gather)
// scx  layout: [sc0(32) | sc1 interleaved u*3+m (96)]        (matches output layout)
__global__ void node_pre_kernel(const float* __restrict__ node_input,
                                const float* __restrict__ node_attr,
                                const float* __restrict__ Wsc0,
                                const float* __restrict__ Wsc1,
                                const float* __restrict__ Wl10,
                                const float* __restrict__ Wl11,
                                float* __restrict__ f0f1,
                                float* __restrict__ scx, int N) {
  int wv = threadIdx.x >> 5;
  int lane = threadIdx.x & 31;
  int n = blockIdx.x * (blockDim.x >> 5) + wv;
  if (n >= N) return;
  const float* xi = node_input + (long)n * 128;
  float f0 = 0.f, f1x = 0.f, f1y = 0.f, f1z = 0.f;
  float s0 = 0.f, s1x = 0.f, s1y = 0.f, s1z = 0.f;
  for (int u = 0; u < 32; ++u) {
    float x0 = xi[u];
    float x1x = xi[32 + u * 3 + 0];
    float x1y = xi[32 + u * 3 + 1];
    float x1z = xi[32 + u * 3 + 2];
    float wl0 = Wl10[u * 32 + lane], wl1 = Wl11[u * 32 + lane];
    float ws0 = Wsc0[u * 32 + lane], ws1 = Wsc1[u * 32 + lane];
    f0 += x0 * wl0;
    f1x += x1x * wl1; f1y += x1y * wl1; f1z += x1z * wl1;
    s0 += x0 * ws0;
    s1x += x1x * ws1; s1y += x1y * ws1; s1z += x1z * ws1;
  }
  float k = node_attr[n] * 0.17677669529663687f;  // 1/sqrt(32)
  float* fo = f0f1 + (long)n * 128;
  fo[lane] = f0 * k;
  fo[32 + lane] = f1x * k;
  fo[64 + lane] = f1y * k;
  fo[96 + lane] = f1z * k;
  float* so = scx + (long)n * 128;
  so[lane] = s0 * k;
  so[32 + lane * 3 + 0] = s1x * k;
  so[32 + lane * 3 + 1] = s1y * k;
  so[32 + lane * 3 + 2] = s1z * k;
}

#define EK_WAVES 8
#define HSTR 68  // H staging row stride (floats): r*68 mod 64 distinct -> no bank conflicts

__global__ void __launch_bounds__(256)
edge_kernel(const float* __restrict__ edge_scalars,
            const float* __restrict__ edge_attr,
            const int* __restrict__ edge_src,
            const int* __restrict__ edge_dst,
            const float* __restrict__ Wfc1,   // (16,64)
            const float* __restrict__ Wfc2,   // (64,128)
            const float* __restrict__ f0f1,
            float* __restrict__ mid, int E) {
  __shared__ float fc1[4 * 4 * 32 * 2];        // 4KB  B-fragments [j][t][lane]{.x,.y}
  __shared__ float fc2[16 * 8 * 32 * 2];       // 32KB B-fragments [j][t2][lane]{.x,.y}
  __shared__ float stage[EK_WAVES][2048];      // 64KB per-wave H/w staging

  // Pre-swizzle weights into WMMA B-fragment order:
  // frag(j,t)[lane] = {W[4j+2*hi][16t+n], W[4j+2*hi+1][16t+n]}, hi=lane>>4, n=lane&15
  for (int idx = threadIdx.x; idx < 4 * 4 * 32; idx += blockDim.x) {
    int ln = idx & 31, t = (idx >> 5) & 3, j = idx >> 7;
    int row = 4 * j + 2 * (ln >> 4), col = 16 * t + (ln & 15);
    fc1[idx * 2 + 0] = Wfc1[row * 64 + col];
    fc1[idx * 2 + 1] = Wfc1[(row + 1) * 64 + col];
  }
  for (int idx = threadIdx.x; idx < 16 * 8 * 32; idx += blockDim.x) {
    int ln = idx & 31, t = (idx >> 5) & 7, j = idx >> 8;
    int row = 4 * j + 2 * (ln >> 4), col = 16 * t + (ln & 15);
    fc2[idx * 2 + 0] = Wfc2[row * 128 + col];
    fc2[idx * 2 + 1] = Wfc2[(row + 1) * 128 + col];
  }
  __syncthreads();

  const int lane = threadIdx.x & 31;
  const int wv = threadIdx.x >> 5;
  const int r = lane & 15;   // A-matrix row (edge within tile) / B column
  const int hi = lane >> 4;  // K-pair selector for f32 WMMA layout
  float* st = stage[wv];
  const int nw = gridDim.x * (blockDim.x >> 5);
  const int gw = blockIdx.x * (blockDim.x >> 5) + wv;
  const int tiles = (E + 15) >> 4;

  for (int tile = gw; tile < tiles; tile += nw) {  // wave-uniform loop: EXEC all-1s for WMMA
    const int e0 = tile * 16;
    int er = e0 + r;
    if (er >= E) er = E - 1;  // clamp row for partial tile (extra rows ignored later)
    const float* srow = edge_scalars + (long)er * 16;

    // A fragments for GEMM1: S tile (16 edges x 16 scalars)
    v2f a1[4];
#pragma unroll
    for (int j = 0; j < 4; ++j) a1[j] = *(const v2f*)(srow + 4 * j + 2 * hi);

    // GEMM1: H = silu(S @ Wfc1 * 0.25); stage H (16x64, row stride HSTR)
#pragma unroll
    for (int t = 0; t < 4; ++t) {
      v8f c = {0.f, 0.f, 0.f, 0.f, 0.f, 0.f, 0.f, 0.f};
#pragma unroll
      for (int j = 0; j < 4; ++j) {
        v2f b = *(const v2f*)&fc1[((j * 4 + t) * 32 + lane) * 2];
        c = wmma4(a1[j], b, c);
      }
#pragma unroll
      for (int v = 0; v < 8; ++v) {
        float x = c[v] * 0.25f;                                 // /sqrt(16)
        st[(v + 8 * hi) * HSTR + 16 * t + r] = x / (1.0f + __expf(-x));  // silu
      }
    }

    // Reload H as A fragments for GEMM2 (K=64 -> 16 fragments)
    v2f a2[16];
#pragma unroll
    for (int j = 0; j < 16; ++j)
      a2[j] = *(const v2f*)&st[r * HSTR + 4 * j + 2 * hi];

    // GEMM2: w = H @ Wfc2 * 0.125; stage w (16x128, row stride 128)
#pragma unroll
    for (int t = 0; t < 8; ++t) {
      v8f c = {0.f, 0.f, 0.f, 0.f, 0.f, 0.f, 0.f, 0.f};
#pragma unroll
      for (int j = 0; j < 16; ++j) {
        v2f b = *(const v2f*)&fc2[((j * 8 + t) * 32 + lane) * 2];
        c = wmma4(a2[j], b, c);
      }
#pragma unroll
      for (int v = 0; v < 8; ++v)
        st[(v + 8 * hi) * 128 + 16 * t + r] = c[v] * 0.125f;    // /sqrt(64)
    }

    // Edge features + coalesced scatter-add. lane = channel u.
    const float q = 0.31622776601683794f;   // 1/sqrt(NUM_NEIGHBORS) folded in
    const float r3 = 0.5773502691896258f;   // 1/sqrt(3)
    for (int i = 0; i < 16; ++i) {
      int e = e0 + i;
      if (e >= E) break;  // uniform branch
      int s = edge_src[e], d = edge_dst[e];
      const float4 ya = *(const float4*)(edge_attr + (long)e * 4);  // y0, y1[3]
      const float* g = f0f1 + (long)s * 128;                        // L2-resident gather
      float g0 = g[lane];
      float g1x = g[32 + lane], g1y = g[64 + lane], g1z = g[96 + lane];
      float wA = st[i * 128 + lane];
      float wB = st[i * 128 + 32 + lane];
      float wC = st[i * 128 + 64 + lane];
      float wD = st[i * 128 + 96 + lane];
      float eA = wA * g0 * ya.x * q;
      float eD = wD * (g1x * ya.y + g1y * ya.z + g1z * ya.w) * r3 * q;
      float gB = wB * g0 * q;
      float cc = wC * ya.x * q;
      float* m = mid + (long)d * 256;
      atomicAdd(m + lane, eA);                       // eA -> mid[:, u]
      atomicAdd(m + 32 + lane, eD);                  // eD -> mid[:, 32+u]
      atomicAdd(m + 64 + 3 * lane + 0, gB * ya.y);   // eB -> mid[:, 64+3u+m]
      atomicAdd(m + 64 + 3 * lane + 1, gB * ya.z);
      atomicAdd(m + 64 + 3 * lane + 2, gB * ya.w);
      atomicAdd(m + 160 + 3 * lane + 0, cc * g1x);   // eC -> mid[:, 160+3u+m]
      atomicAdd(m + 160 + 3 * lane + 1, cc * g1y);
      atomicAdd(m + 160 + 3 * lane + 2, cc * g1z);
    }
  }
}

// One wave per node; lane = output channel w.
__global__ void node_post_kernel(const float* __restrict__ mid,
                                 const float* __restrict__ node_attr,
                                 const float* __restrict__ Wl20,
                                 const float* __restrict__ Wl21,
                                 const float* __restrict__ Wl3,
                                 const float* __restrict__ scx,
                                 float* __restrict__ out, int N) {
  int wv = threadIdx.x >> 5;
  int lane = threadIdx.x & 31;
  int n = blockIdx.x * (blockDim.x >> 5) + wv;
  if (n >= N) return;
  const float* m = mid + (long)n * 256;
  float o0 = 0.f, o1x = 0.f, o1y = 0.f, o1z = 0.f, ang = 0.f;
  for (int v = 0; v < 64; ++v) {
    float m0 = m[v];
    float mx = m[64 + v * 3 + 0];
    float my = m[64 + v * 3 + 1];
    float mz = m[64 + v * 3 + 2];
    float w0 = Wl20[v * 32 + lane], w1 = Wl21[v * 32 + lane];
    o0 += m0 * w0;
    o1x += mx * w1; o1y += my * w1; o1z += mz * w1;
    ang += m0 * Wl3[v];
  }
  float k = node_attr[n] * 0.125f;  // inv8
  o0 *= k; o1x *= k; o1y *= k; o1z *= k;
  float angle = 0.1f * ang * k;
  float ca = cosf(angle), sa = sinf(angle);
  const float* sx = scx + (long)n * 128;
  float* o = out + (long)n * 128;
  o[lane] = ca * sx[lane] + sa * o0;
  o[32 + 3 * lane + 0] = ca * sx[32 + 3 * lane + 0] + sa * o1x;
  o[32 + 3 * lane + 1] = ca * sx[32 + 3 * lane + 1] + sa * o1y;
  o[32 + 3 * lane + 2] = ca * sx[32 + 3 * lane + 2] + sa * o1z;
}

extern "C" void kernel_launch(void* const* d_in, const int* in_sizes, int n_in,
                              void* d_out, int out_size, void* d_ws, size_t ws_size,
                              hipStream_t stream) {
  const float* node_input   = (const float*)d_in[0];
  const float* node_attr    = (const float*)d_in[1];
  const int*   edge_src     = (const int*)d_in[2];
  const int*   edge_dst     = (const int*)d_in[3];
  const float* edge_attr    = (const float*)d_in[4];
  const float* edge_scalars = (const float*)d_in[5];
  const float* W_sc0        = (const float*)d_in[6];
  const float* W_sc1        = (const float*)d_in[7];
  const float* W_lin1_0     = (const float*)d_in[8];
  const float* W_lin1_1     = (const float*)d_in[9];
  const float* W_fc1        = (const float*)d_in[10];
  const float* W_fc2        = (const float*)d_in[11];
  const float* W_lin2_0     = (const float*)d_in[12];
  const float* W_lin2_1     = (const float*)d_in[13];
  const float* W_lin3       = (const float*)d_in[14];
  int N = in_sizes[1];
  int E = in_sizes[2];

  float* f0f1 = (float*)d_ws;
  float* scx  = f0f1 + (size_t)N * 128;
  float* mid  = scx + (size_t)N * 128;
  float* out  = (float*)d_out;

  zero_kernel<<<2048, 256, 0, stream>>>(mid, (long)N * 256);

  int nblk_node = (N + 7) / 8;  // 8 waves/block, wave-per-node
  node_pre_kernel<<<nblk_node, 256, 0, stream>>>(
      node_input, node_attr, W_sc0, W_sc1, W_lin1_0, W_lin1_1, f0f1, scx, N);

  edge_kernel<<<1024, 256, 0, stream>>>(
      edge_scalars, edge_attr, edge_src, edge_dst, W_fc1, W_fc2, f0f1, mid, E);

  node_post_kernel<<<nblk_node, 256, 0, stream>>>(
      mid, node_attr, W_lin2_0, W_lin2_1, W_lin3, scx, out, N);
}